// LSTM_42039139894219
// MI455X (gfx1250) — compile-verified
//
#include <hip/hip_runtime.h>

#define SEQ   33
#define FDIM  20480
#define HN    900          // H1 == H2
#define BN    2
#define WVN   100
#define OBJN  35
#define RELN  82
#define G4    3600         // 4*H
#define MROWS 66           // SEQ*BN

typedef __attribute__((ext_vector_type(2))) float v2f;
typedef __attribute__((ext_vector_type(8))) float v8f;

__device__ __forceinline__ float sigf(float x) { return 1.0f / (1.0f + __expf(-x)); }

// ---------------- init kernels ----------------
__global__ void k_init_bias1(const float* __restrict__ bi, const float* __restrict__ bh,
                             float* __restrict__ pre) {
    int i = blockIdx.x * blockDim.x + threadIdx.x;           // over 66*3600
    if (i >= MROWS * G4) return;
    int j = i % G4;
    pre[i] = bi[j] + bh[j];
}

__global__ void k_init_bias2(const float* __restrict__ bi, const float* __restrict__ bh,
                             float* __restrict__ pre) {
    int i = blockIdx.x * blockDim.x + threadIdx.x;           // over 66*3600
    if (i >= MROWS * G4) return;
    int j = i % G4;
    int t = (i / G4) >> 1;
    pre[i] = bi[(long)t * G4 + j] + bh[(long)t * G4 + j];
}

__global__ void k_zero(float* __restrict__ p, int n) {
    int i = blockIdx.x * blockDim.x + threadIdx.x;
    if (i < n) p[i] = 0.0f;
}

// ---------------- big GEMM 1: preact1 += X[66,20480] @ W[3600,20480]^T ----------------
// grid (225, 16), block 32 (one wave). K-chunk = 1280, 320 iterations of k-step 4.
__global__ __launch_bounds__(32)
void k_gemm1(const float* __restrict__ X, const float* __restrict__ W,
             float* __restrict__ Out) {
    const int lane = threadIdx.x;
    const int lo = lane & 15, hi = lane >> 4;
    const int n = blockIdx.x * 16 + lo;                      // output column
    const int kbeg = blockIdx.y * (FDIM / 16);
    const float m4 = (lo < 2) ? 1.0f : 0.0f;                 // tile-4 valid-row mask
    const int r4 = 64 + (lo & 1);                            // always-valid clamped row

    const float* wp = W + (long)n * FDIM + kbeg + 2 * hi;    // B frag stream (read once)
    const float* xp = X + kbeg + 2 * hi;                     // A frag base (L2 resident)

    v8f acc0 = {0,0,0,0,0,0,0,0}, acc1 = acc0, acc2 = acc0, acc3 = acc0, acc4 = acc0;

    for (int it = 0; it < (FDIM / 16) / 4; ++it) {
        v2f bf = __builtin_nontemporal_load((const v2f*)wp);
        v2f a0 = *(const v2f*)(xp + (long)(lo)      * FDIM);
        v2f a1 = *(const v2f*)(xp + (long)(16 + lo) * FDIM);
        v2f a2 = *(const v2f*)(xp + (long)(32 + lo) * FDIM);
        v2f a3 = *(const v2f*)(xp + (long)(48 + lo) * FDIM);
        v2f a4 = *(const v2f*)(xp + (long)(r4)      * FDIM);
        a4 *= m4;                                            // branchless zero-fill
        acc0 = __builtin_amdgcn_wmma_f32_16x16x4_f32(false, a0, false, bf, (short)0, acc0, false, false);
        acc1 = __builtin_amdgcn_wmma_f32_16x16x4_f32(false, a1, false, bf, (short)0, acc1, false, false);
        acc2 = __builtin_amdgcn_wmma_f32_16x16x4_f32(false, a2, false, bf, (short)0, acc2, false, false);
        acc3 = __builtin_amdgcn_wmma_f32_16x16x4_f32(false, a3, false, bf, (short)0, acc3, false, false);
        acc4 = __builtin_amdgcn_wmma_f32_16x16x4_f32(false, a4, false, bf, (short)0, acc4, false, false);
        wp += 4;
        xp += 4;
    }
    #pragma unroll
    for (int r = 0; r < 8; ++r) {
        int m = r + 8 * hi;                                  // row within tile
        atomicAdd(&Out[(long)(m)      * G4 + n], acc0[r]);
        atomicAdd(&Out[(long)(16 + m) * G4 + n], acc1[r]);
        atomicAdd(&Out[(long)(32 + m) * G4 + n], acc2[r]);
        atomicAdd(&Out[(long)(48 + m) * G4 + n], acc3[r]);
        if (64 + m < MROWS) atomicAdd(&Out[(long)(64 + m) * G4 + n], acc4[r]);
    }
}

// ------------- big GEMM 2: preact2[t] += out1[t][2,900] @ w_ih2[t][:, :900]^T -------------
// grid (225, 5, 33), block 32. K-chunk = 180, 45 iterations.
__global__ __launch_bounds__(32)
void k_gemm2(const float* __restrict__ out1, const float* __restrict__ Wih2,
             float* __restrict__ pre2) {
    const int lane = threadIdx.x;
    const int lo = lane & 15, hi = lane >> 4;
    const int n = blockIdx.x * 16 + lo;
    const int t = blockIdx.z;
    const int kbeg = blockIdx.y * 180;
    const float mA = (lo < BN) ? 1.0f : 0.0f;

    const float* ap = out1 + (long)t * BN * HN + (long)(lo & 1) * HN + kbeg + 2 * hi;
    const float* wp = Wih2 + (long)t * G4 * 1000 + (long)n * 1000 + kbeg + 2 * hi;

    v8f acc = {0,0,0,0,0,0,0,0};
    for (int it = 0; it < 180 / 4; ++it) {
        v2f bf = __builtin_nontemporal_load((const v2f*)wp);
        v2f af = *(const v2f*)ap;
        af *= mA;                                            // branchless zero-fill rows >= 2
        acc = __builtin_amdgcn_wmma_f32_16x16x4_f32(false, af, false, bf, (short)0, acc, false, false);
        wp += 4;
        ap += 4;
    }
    if (hi == 0) {                                           // only M rows 0,1 valid
        float* dst = pre2 + (long)t * BN * G4;
        atomicAdd(&dst[n],      acc[0]);
        atomicAdd(&dst[G4 + n], acc[1]);
    }
}

// ------------- fused recurrent step: one wave per hidden unit j -------------
// gates[b][g] = pre_t[b][g*900+j] + hin[b] . Whh[g*900+j]; then LSTM cell.
// hin is read-only (prev step's h), hout written; c updated in place (owner-exclusive).
// grid 225 blocks x 128 threads (4 waves) -> j = 0..899.
__global__ __launch_bounds__(128)
void k_step(const float* __restrict__ pre_t,   // [2,3600]
            const float* __restrict__ Whh,     // [3600,900]
            const float* __restrict__ hin,     // [2,900]
            float* __restrict__ c,             // [2,900] in place
            float* __restrict__ hout) {        // [2,900]
    const int wave = threadIdx.x >> 5;
    const int lane = threadIdx.x & 31;
    const int j = blockIdx.x * 4 + wave;       // 0..899

    float a[8] = {0,0,0,0,0,0,0,0};            // [gate*2 + b]
    for (int k = lane; k < HN; k += 32) {      // coalesced 128B lines per load
        float h0 = hin[k];
        float h1 = hin[HN + k];
        #pragma unroll
        for (int g = 0; g < 4; ++g) {
            float w = Whh[(long)(g * HN + j) * HN + k];
            a[2 * g]     = fmaf(w, h0, a[2 * g]);
            a[2 * g + 1] = fmaf(w, h1, a[2 * g + 1]);
        }
    }
    #pragma unroll
    for (int off = 16; off >= 1; off >>= 1) {
        #pragma unroll
        for (int q = 0; q < 8; ++q) a[q] += __shfl_xor(a[q], off, 32);
    }
    if (lane == 0) {
        #pragma unroll
        for (int b = 0; b < BN; ++b) {
            const float* g = pre_t + (long)b * G4;
            float ig = sigf(g[j]               + a[0 + b]);
            float fg = sigf(g[HN + j]          + a[2 + b]);
            float gg = tanhf(g[2 * HN + j]     + a[4 + b]);
            float og = sigf(g[3 * HN + j]      + a[6 + b]);
            float cc = fg * c[b * HN + j] + ig * gg;
            float hh = og * tanhf(cc);
            c[b * HN + j] = cc;
            hout[b * HN + j] = hh;
        }
    }
}

// ------------- word-vector columns for t = 31, 32: gates += wv @ Wih2[t][:, 900:1000]^T -------------
__global__ void k_wvgates(const float* __restrict__ wv, const float* __restrict__ Wih2_t,
                          float* __restrict__ gates) {
    int jj = blockIdx.x * blockDim.x + threadIdx.x;
    if (jj >= G4) return;
    const float* wr = Wih2_t + (long)jj * 1000 + HN;
    float p0 = 0.0f, p1 = 0.0f;
    for (int k = 0; k < WVN; ++k) {
        float w = wr[k];
        p0 = fmaf(w, wv[k], p0);
        p1 = fmaf(w, wv[WVN + k], p1);
    }
    gates[jj]      += p0;
    gates[G4 + jj] += p1;
}

// ------------- classifier heads: out[b][o] = bias[o] + h2[b].W[o] -------------
__global__ void k_head(const float* __restrict__ h2, const float* __restrict__ W,
                       const float* __restrict__ bias, float* __restrict__ out, int N) {
    int i = blockIdx.x * blockDim.x + threadIdx.x;
    if (i >= BN * N) return;
    int b = i / N, o = i % N;
    const float* wr = W + (long)o * HN;
    const float* hb = h2 + (long)b * HN;
    float s = bias[o];
    for (int k = 0; k < HN; ++k) s = fmaf(wr[k], hb[k], s);
    out[(long)b * N + o] = s;
}

// ------------- argmax over logits + embedding gather -------------
__global__ void k_argmax_embed(const float* __restrict__ logits, int N, int base,
                               const float* __restrict__ embed, float* __restrict__ wv) {
    __shared__ int idx[BN];
    int tid = threadIdx.x;
    if (tid < BN) {
        const float* l = logits + (long)tid * N;
        int best = 0; float bv = l[0];
        for (int k = 1; k < N; ++k)
            if (l[k] > bv) { bv = l[k]; best = k; }          // strict > keeps first max
        idx[tid] = best + base;
    }
    __syncthreads();
    for (int i = tid; i < BN * WVN; i += blockDim.x) {
        int b = i / WVN, w = i % WVN;
        wv[i] = embed[(long)idx[b] * WVN + w];
    }
}

extern "C" void kernel_launch(void* const* d_in, const int* in_sizes, int n_in,
                              void* d_out, int out_size, void* d_ws, size_t ws_size,
                              hipStream_t stream) {
    const float* x      = (const float*)d_in[0];
    const float* w_ih1  = (const float*)d_in[1];
    const float* w_hh1  = (const float*)d_in[2];
    const float* b_ih1  = (const float*)d_in[3];
    const float* b_hh1  = (const float*)d_in[4];
    const float* w_ih2  = (const float*)d_in[5];
    const float* w_hh2  = (const float*)d_in[6];
    const float* b_ih2  = (const float*)d_in[7];
    const float* b_hh2  = (const float*)d_in[8];
    const float* w_obj1 = (const float*)d_in[9];
    const float* b_obj1 = (const float*)d_in[10];
    const float* w_rel  = (const float*)d_in[11];
    const float* b_rel  = (const float*)d_in[12];
    const float* w_obj2 = (const float*)d_in[13];
    const float* b_obj2 = (const float*)d_in[14];
    const float* embed  = (const float*)d_in[15];

    float* out = (float*)d_out;                  // [2,35] | [2,82] | [2,35] = 304 floats
    float* ws = (float*)d_ws;
    float* preact1 = ws;                         // 66*3600
    float* out1    = preact1 + (long)MROWS * G4; // 66*900
    float* preact2 = out1 + (long)MROWS * HN;    // 66*3600
    float* h1z = preact2 + (long)MROWS * G4;     // 1800 (zero h for layer-1 t=0)
    float* c1  = h1z + BN * HN;                  // 1800
    float* h2A = c1 + BN * HN;                   // 1800 (ping)
    float* h2B = h2A + BN * HN;                  // 1800 (pong)
    float* c2  = h2B + BN * HN;                  // 1800
    float* wv  = c2 + BN * HN;                   // 200

    const int npre = MROWS * G4;
    k_init_bias1<<<(npre + 255) / 256, 256, 0, stream>>>(b_ih1, b_hh1, preact1);
    k_init_bias2<<<(npre + 255) / 256, 256, 0, stream>>>(b_ih2, b_hh2, preact2);
    k_zero<<<(5 * BN * HN + BN * WVN + 255) / 256, 256, 0, stream>>>(h1z, 5 * BN * HN + BN * WVN);

    // layer-1 input GEMM (streams 295 MB once)
    k_gemm1<<<dim3(G4 / 16, 16), 32, 0, stream>>>(x, w_ih1, preact1);

    // layer-1 recurrence: read out1[t-1], write out1[t] (no h race)
    for (int t = 0; t < SEQ; ++t) {
        const float* hin = (t == 0) ? h1z : out1 + (long)(t - 1) * BN * HN;
        k_step<<<HN / 4, 128, 0, stream>>>(preact1 + (long)t * BN * G4, w_hh1,
                                           hin, c1, out1 + (long)t * BN * HN);
    }

    // layer-2 input GEMM for all t (streams ~428 MB once)
    k_gemm2<<<dim3(G4 / 16, 5, SEQ), 32, 0, stream>>>(out1, w_ih2, preact2);

    // layer-2 recurrence with h ping-pong; c2 in place
    float* hc = h2A;                             // zeroed
    float* hn = h2B;
    for (int t = 0; t < SEQ - 2; ++t) {
        k_step<<<HN / 4, 128, 0, stream>>>(preact2 + (long)t * BN * G4,
                                           w_hh2 + (long)t * G4 * HN, hc, c2, hn);
        float* tmp = hc; hc = hn; hn = tmp;
    }

    // obj1 head -> argmax -> wv -> step SEQ-2
    k_head<<<1, 128, 0, stream>>>(hc, w_obj1, b_obj1, out, OBJN);
    k_argmax_embed<<<1, 128, 0, stream>>>(out, OBJN, 0, embed, wv);
    {
        int t = SEQ - 2;
        float* g = preact2 + (long)t * BN * G4;
        k_wvgates<<<(G4 + 255) / 256, 256, 0, stream>>>(wv, w_ih2 + (long)t * G4 * 1000, g);
        k_step<<<HN / 4, 128, 0, stream>>>(g, w_hh2 + (long)t * G4 * HN, hc, c2, hn);
        float* tmp = hc; hc = hn; hn = tmp;
    }

    // rel head -> argmax(+OBJ) -> wv -> step SEQ-1
    k_head<<<1, 192, 0, stream>>>(hc, w_rel, b_rel, out + BN * OBJN, RELN);
    k_argmax_embed<<<1, 128, 0, stream>>>(out + BN * OBJN, RELN, OBJN, embed, wv);
    {
        int t = SEQ - 1;
        float* g = preact2 + (long)t * BN * G4;
        k_wvgates<<<(G4 + 255) / 256, 256, 0, stream>>>(wv, w_ih2 + (long)t * G4 * 1000, g);
        k_step<<<HN / 4, 128, 0, stream>>>(g, w_hh2 + (long)t * G4 * HN, hc, c2, hn);
        float* tmp = hc; hc = hn; hn = tmp;
    }

    // obj2 head
    k_head<<<1, 128, 0, stream>>>(hc, w_obj2, b_obj2, out + BN * OBJN + BN * RELN, OBJN);
}